// Threshold_weights_26147760898280
// MI455X (gfx1250) — compile-verified
//
#include <hip/hip_runtime.h>
#include <math.h>
#include <stdint.h>

// ---------------------------------------------------------------------------
// Workload constants (match reference)
// ---------------------------------------------------------------------------
#define N_ROWS     16384
#define N_COLS     1000
#define N_VEC      250           // float4 chunks per row (1000/4)
#define N_VEC_PAD  256           // padded to 8 full wave32 chunks
#define INV_TEMP   0.5f          // 1 / TEMPERATURE (T = 2.0)

typedef float v4f __attribute__((ext_vector_type(4)));
typedef int   v4i __attribute__((ext_vector_type(4)));

// ---------------------------------------------------------------------------
// gfx1250 async global->LDS copy (ASYNCcnt-tracked). The builtin's params are
// int4* in device/shared address spaces (leaked by the round-2 diagnostic);
// pass generic-pointer casts with the exact pointee type.
// ---------------------------------------------------------------------------
__device__ __forceinline__ void async_load_b128(const void* gsrc, void* ldst) {
#if __has_builtin(__builtin_amdgcn_global_load_async_to_lds_b128)
    __builtin_amdgcn_global_load_async_to_lds_b128(
        (v4i*)gsrc, (v4i*)ldst, /*imm offset*/ 0, /*cpol*/ 0);
#else
    // AS(3) pointer value == in-LDS byte offset (low 32 bits of flat addr).
    asm volatile("global_load_async_to_lds_b128 %0, %1, off"
                 :
                 : "v"((uint32_t)(uintptr_t)ldst),
                   "v"((uint64_t)(uintptr_t)gsrc)
                 : "memory");
#endif
}

__device__ __forceinline__ void wait_async0() {
#if __has_builtin(__builtin_amdgcn_s_wait_asynccnt)
    __builtin_amdgcn_s_wait_asynccnt(0);
#else
    asm volatile("s_wait_asynccnt 0x0" ::: "memory");
#endif
    asm volatile("" ::: "memory");   // keep LDS reads below the wait
}

// Float atomic-max via integer atomics (monotone bit tricks); correct for
// mixed signs given destination initialized to -inf.
__device__ __forceinline__ void atomicMaxFloat(float* addr, float v) {
    if (v >= 0.0f) {
        atomicMax((int*)addr, __float_as_int(v));
    } else {
        atomicMin((unsigned int*)addr, __float_as_uint(v));
    }
}

__global__ void init_out_kernel(float* __restrict__ out) {
    out[0] = -INFINITY;   // identity for the global max reduction
}

// One block per row; 5 wave32s per block, wave w reduces matrix w's row.
__global__ __launch_bounds__(160) void margin_softmax_kernel(
    const float* __restrict__ o1, const float* __restrict__ o2,
    const float* __restrict__ o3, const float* __restrict__ o4,
    const float* __restrict__ mimic, const int* __restrict__ targets,
    float* __restrict__ out)
{
    __shared__ v4f   tile[5][N_VEC_PAD];   // 20480 B staged rows
    __shared__ float sMargin[5];
    __shared__ float sMax[4];

    const int row  = blockIdx.x;
    const int wave = threadIdx.x >> 5;   // 0..4 -> which matrix
    const int lane = threadIdx.x & 31;

    const float* base =
        (wave == 0) ? o1 : (wave == 1) ? o2 : (wave == 2) ? o3 :
        (wave == 3) ? o4 : mimic;
    const v4f* __restrict__ rowp = (const v4f*)(base + (size_t)row * N_COLS);

    // Fire the whole 4000 B row as 8 async 512 B wave transfers into LDS.
    // No VGPR return operands -> low register pressure, ASYNCcnt-tracked.
    // Tail chunks clamp the SOURCE index (stay in-bounds) but still fill
    // padded LDS slots 250..255, which are never read back.
    #pragma unroll
    for (int k = 0; k < 8; ++k) {
        const int i    = lane + (k << 5);
        const int isrc = (i < N_VEC) ? i : (N_VEC - 1);
        async_load_b128(rowp + isrc, &tile[wave][i]);
    }
    wait_async0();

    // Target value: single broadcast LDS read by all lanes of this wave.
    const int   tgt = targets[row];
    const float tv  = ((const float*)tile[wave])[tgt];

    // Per-lane top-2 over staged data (duplicate-safe update).
    float m1 = -INFINITY, m2 = -INFINITY;
    #pragma unroll
    for (int k = 0; k < 8; ++k) {
        const int i = lane + (k << 5);
        if (i < N_VEC) {
            const v4f v = tile[wave][i];
            float e;
            e = v.x; if (e > m1) { m2 = m1; m1 = e; } else if (e > m2) { m2 = e; }
            e = v.y; if (e > m1) { m2 = m1; m1 = e; } else if (e > m2) { m2 = e; }
            e = v.z; if (e > m1) { m2 = m1; m1 = e; } else if (e > m2) { m2 = e; }
            e = v.w; if (e > m1) { m2 = m1; m1 = e; } else if (e > m2) { m2 = e; }
        }
    }

    // wave32 butterfly merge of (top1, top2) pairs:
    //   m1' = max(a1,b1); m2' = max(min(a1,b1), max(a2,b2))
    #pragma unroll
    for (int off = 16; off > 0; off >>= 1) {
        const float a1 = __shfl_xor(m1, off, 32);
        const float a2 = __shfl_xor(m2, off, 32);
        const float hi = fmaxf(m1, a1);
        const float lo = fminf(m1, a1);
        m2 = fmaxf(lo, fmaxf(m2, a2));
        m1 = hi;
    }

    if (lane == 0) {
        sMargin[wave] = (m1 == tv) ? (m1 - m2) : 0.0f;
        if (wave < 4) sMax[wave] = m1;
    }
    __syncthreads();

    if (threadIdx.x == 0) {
        // 5-way stable softmax of margins / T
        float x[5], e[5];
        float xm = -INFINITY;
        #pragma unroll
        for (int j = 0; j < 5; ++j) {
            x[j] = sMargin[j] * INV_TEMP;
            xm = fmaxf(xm, x[j]);
        }
        float s = 0.0f;
        #pragma unroll
        for (int j = 0; j < 5; ++j) {
            e[j] = expf(x[j] - xm);
            s += e[j];
        }
        const float inv = 1.0f / s;
        float* orow = out + 1 + (size_t)row * 5;   // out[0] = scalar max_preds
        #pragma unroll
        for (int j = 0; j < 5; ++j)
            orow[j] = e[j] * inv;

        // global max over matrices 1..4 (mimic excluded)
        const float bm = fmaxf(fmaxf(sMax[0], sMax[1]), fmaxf(sMax[2], sMax[3]));
        atomicMaxFloat(out, bm);
    }
}

extern "C" void kernel_launch(void* const* d_in, const int* in_sizes, int n_in,
                              void* d_out, int out_size, void* d_ws, size_t ws_size,
                              hipStream_t stream) {
    const float* o1    = (const float*)d_in[0];
    const float* o2    = (const float*)d_in[1];
    const float* o3    = (const float*)d_in[2];
    const float* o4    = (const float*)d_in[3];
    const float* mimic = (const float*)d_in[4];
    const int*   tgts  = (const int*)d_in[5];
    float* out = (float*)d_out;

    (void)in_sizes; (void)n_in; (void)out_size; (void)d_ws; (void)ws_size;

    // Initialize the scalar-max slot to -inf before atomics (same stream ->
    // ordered; graph-capture safe, no sync APIs).
    init_out_kernel<<<1, 1, 0, stream>>>(out);
    margin_softmax_kernel<<<N_ROWS, 160, 0, stream>>>(o1, o2, o3, o4, mimic,
                                                      tgts, out);
}